// MultiDecoder_4131758539248
// MI455X (gfx1250) — compile-verified
//
#include <hip/hip_runtime.h>
#include <cmath>

typedef __attribute__((ext_vector_type(2))) float v2f;
typedef __attribute__((ext_vector_type(4))) float v4f;
typedef __attribute__((ext_vector_type(8))) float v8f;

#define NN   8192
#define DD   64
#define LL   4
#define FF   128

// ---------------------------------------------------------------------------
// Pre-kernel: sq[n] = sum_d Z[n,d]^2        (8192 floats,  ws offset 0)
//             ybeta[l*N+n] = Y[l,n,:]·betas[l,:]  (32768 floats, ws offset 8192)
// ---------------------------------------------------------------------------
__global__ __launch_bounds__(256) void precompute_kernel(
    const float* __restrict__ Z, const float* __restrict__ Y,
    const float* __restrict__ betas, float* __restrict__ ws)
{
    int tid = blockIdx.x * blockDim.x + threadIdx.x;
    if (tid < NN) {
        // sq[n]
        const v4f* z = (const v4f*)(Z + (size_t)tid * DD);
        float s = 0.f;
#pragma unroll
        for (int i = 0; i < DD / 4; ++i) {
            v4f v = z[i];
            s += v.x * v.x + v.y * v.y + v.z * v.z + v.w * v.w;
        }
        ws[tid] = s;
    } else if (tid < NN + LL * NN) {
        int t = tid - NN;
        int l = t >> 13;           // / 8192
        int n = t & (NN - 1);
        const v4f* y = (const v4f*)(Y + ((size_t)l * NN + n) * FF);
        const v4f* b = (const v4f*)(betas + (size_t)l * FF);
        float s = 0.f;
#pragma unroll
        for (int i = 0; i < FF / 4; ++i) {
            v4f yv = y[i];
            v4f bv = b[i];
            s += yv.x * bv.x + yv.y * bv.y + yv.z * bv.z + yv.w * bv.w;
        }
        ws[NN + t] = s;
    }
}

// ---------------------------------------------------------------------------
// Fused kernel: per wave, 16x64 tile of inner = Z·Z^T via V_WMMA_F32_16X16X4_F32,
// then dist + 4-layer sigmoid epilogue streamed to d_out.
// Block: 256 threads = 8 waves -> 128 rows x 64 cols per block.
// ---------------------------------------------------------------------------
__global__ __launch_bounds__(256) void multidecoder_kernel(
    const float* __restrict__ Z,
    const float* __restrict__ alphas,
    const float* __restrict__ gammas,
    const float* __restrict__ ws,      // sq | ybeta
    float* __restrict__ out)
{
    const int lane = threadIdx.x & 31;
    const int wave = threadIdx.x >> 5;
    const int hi   = lane >> 4;        // 0: K={0,1}, 1: K={2,3}
    const int lo   = lane & 15;        // row (A) / col (B) index within tile

    const int colBase = blockIdx.x * 64;
    const int rowBase = blockIdx.y * 128 + wave * 16;

    const float* __restrict__ sq    = ws;
    const float* __restrict__ ybeta = ws + NN;

    // Operand base pointers (one 16-row A block, four 16-col B blocks)
    const float* za  = Z + (size_t)(rowBase + lo)      * DD + 2 * hi;
    const float* zb0 = Z + (size_t)(colBase + lo)      * DD + 2 * hi;
    const float* zb1 = Z + (size_t)(colBase + 16 + lo) * DD + 2 * hi;
    const float* zb2 = Z + (size_t)(colBase + 32 + lo) * DD + 2 * hi;
    const float* zb3 = Z + (size_t)(colBase + 48 + lo) * DD + 2 * hi;

    v8f acc0 = {}, acc1 = {}, acc2 = {}, acc3 = {};

#pragma unroll
    for (int kb = 0; kb < DD; kb += 4) {
        v2f a  = *(const v2f*)(za  + kb);
        v2f b0 = *(const v2f*)(zb0 + kb);
        v2f b1 = *(const v2f*)(zb1 + kb);
        v2f b2 = *(const v2f*)(zb2 + kb);
        v2f b3 = *(const v2f*)(zb3 + kb);
        acc0 = __builtin_amdgcn_wmma_f32_16x16x4_f32(false, a, false, b0,
                                                     (short)0, acc0, false, false);
        acc1 = __builtin_amdgcn_wmma_f32_16x16x4_f32(false, a, false, b1,
                                                     (short)0, acc1, false, false);
        acc2 = __builtin_amdgcn_wmma_f32_16x16x4_f32(false, a, false, b2,
                                                     (short)0, acc2, false, false);
        acc3 = __builtin_amdgcn_wmma_f32_16x16x4_f32(false, a, false, b3,
                                                     (short)0, acc3, false, false);
    }

    // ---- epilogue ---------------------------------------------------------
    // C layout: VGPR vg -> M = vg + 8*hi ; lane lo -> N = lo
    float sqr[8];
#pragma unroll
    for (int vg = 0; vg < 8; ++vg) sqr[vg] = sq[rowBase + vg + 8 * hi];

    float sqc[4], yb[LL][4], al[LL], ga[LL];
#pragma unroll
    for (int t = 0; t < 4; ++t) sqc[t] = sq[colBase + 16 * t + lo];
#pragma unroll
    for (int l = 0; l < LL; ++l) {
        al[l] = alphas[l];
        ga[l] = gammas[l];
#pragma unroll
        for (int t = 0; t < 4; ++t)
            yb[l][t] = ybeta[(size_t)l * NN + colBase + 16 * t + lo];
    }

    v8f accs[4] = {acc0, acc1, acc2, acc3};
#pragma unroll
    for (int t = 0; t < 4; ++t) {
        const size_t colIdx = (size_t)(colBase + 16 * t + lo);
#pragma unroll
        for (int vg = 0; vg < 8; ++vg) {
            float dist = sqr[vg] - 2.0f * accs[t][vg] + sqc[t];
            size_t base = (size_t)(rowBase + vg + 8 * hi) * NN + colIdx;
#pragma unroll
            for (int l = 0; l < LL; ++l) {
                float x = al[l] + yb[l][t] - ga[l] * dist;
                float e = __expf(-x);                       // v_exp_f32
                out[(size_t)l * NN * NN + base] =
                    __builtin_amdgcn_rcpf(1.0f + e);        // v_rcp_f32
            }
        }
    }
}

// ---------------------------------------------------------------------------
extern "C" void kernel_launch(void* const* d_in, const int* in_sizes, int n_in,
                              void* d_out, int out_size, void* d_ws, size_t ws_size,
                              hipStream_t stream)
{
    const float* Z      = (const float*)d_in[0];   // [N, D]
    const float* Y      = (const float*)d_in[1];   // [L, N, F]
    const float* alphas = (const float*)d_in[2];   // [L]
    const float* betas  = (const float*)d_in[3];   // [L, F]
    const float* gammas = (const float*)d_in[4];   // [L]
    float* out = (float*)d_out;                    // [L, N, N]
    float* ws  = (float*)d_ws;                     // sq(8192) | ybeta(32768)

    // Precompute sq and Y_beta
    {
        int total = NN + LL * NN;                  // 40960 threads
        int blocks = (total + 255) / 256;
        precompute_kernel<<<blocks, 256, 0, stream>>>(Z, Y, betas, ws);
    }

    // Fused dist + sigmoid stream
    {
        dim3 grid(NN / 64, NN / 128);              // (128, 64)
        multidecoder_kernel<<<grid, dim3(256), 0, stream>>>(Z, alphas, gammas, ws, out);
    }
}